// BayesianGCN_13228499272210
// MI455X (gfx1250) — compile-verified
//
#include <hip/hip_runtime.h>
#include <hip/hip_bf16.h>
#include <math.h>

typedef float v2f __attribute__((ext_vector_type(2)));
typedef float v8f __attribute__((ext_vector_type(8)));

namespace {
constexpr int kN = 100000;   // nodes
constexpr int kE = 1600000;  // edges
constexpr int kF = 256;      // input features
constexpr int kH = 128;      // hidden
constexpr int kC = 16;       // classes
constexpr int MT = 32;       // GEMM M-tile rows per block (100000 % 32 == 0 -> 3125 blocks)
constexpr int LDX = kF + 4;  // padded LDS row stride (floats): 260 % 64 == 4 -> conflict-free
}

__device__ __forceinline__ void atomAddF(float* p, float v) {
  __hip_atomic_fetch_add(p, v, __ATOMIC_RELAXED, __HIP_MEMORY_SCOPE_AGENT);
}

// ---------------- degree / normalization -------------------------------------
__global__ void k_deg_init(float* __restrict__ deg) {
  int i = blockIdx.x * blockDim.x + threadIdx.x;
  if (i < kN) deg[i] = 1.0f;  // self-loop contributes 1 to every node's degree
}

__global__ void k_deg_count(const int* __restrict__ ei, float* __restrict__ deg) {
  int e = blockIdx.x * blockDim.x + threadIdx.x;
  if (e < kE) atomAddF(&deg[ei[kE + e]], 1.0f);  // dst row of edge_index
}

__global__ void k_deg_rsqrt(float* __restrict__ deg) {
  int i = blockIdx.x * blockDim.x + threadIdx.x;
  if (i < kN) deg[i] = rsqrtf(deg[i]);  // deg >= 1 always (self-loops)
}

// ---------------- GEMM h = x @ W via V_WMMA_F32_16X16X4_F32 ------------------
// Block: 256 threads = 8 waves. Each wave: one 16-wide N-tile (8*16 = H),
// two 16-row M-subtiles (MT = 32). A staged in LDS, B streamed from L0/L2.
__global__ __launch_bounds__(256) void k_gemm_xw(const float* __restrict__ x,
                                                 const float* __restrict__ W,
                                                 float* __restrict__ h) {
  __shared__ float xs[MT * LDX];
  const int tid = threadIdx.x;
  const int rb  = blockIdx.x * MT;

  // Cooperative load of the 32x256 x-tile: 2048 float4, 8 per thread.
#pragma unroll
  for (int j = 0; j < (MT * kF / 4) / 256; ++j) {
    int idx = tid + j * 256;        // float4 index within tile
    int row = idx / (kF / 4);       // 0..31
    int c4  = idx % (kF / 4);       // 0..63
    float4 v = *(const float4*)(x + (size_t)(rb + row) * kF + c4 * 4);
    float* d = &xs[row * LDX + c4 * 4];
    d[0] = v.x; d[1] = v.y; d[2] = v.z; d[3] = v.w;
  }
  __syncthreads();

  const int wave = tid >> 5;
  const int lane = tid & 31;
  const int hf   = lane >> 4;   // which 16-lane half
  const int r    = lane & 15;
  const int n0   = wave * 16;

  v8f acc0 = {};  // rows rb .. rb+15
  v8f acc1 = {};  // rows rb+16 .. rb+31

#pragma unroll 4
  for (int k = 0; k < kF; k += 4) {
    // B fragment (4x16, K split across lane halves: lanes0-15 K={k,k+1},
    // lanes16-31 K={k+2,k+3}; N = lane%16)
    v2f b;
    b.x = W[(size_t)(k + 2 * hf) * kH + n0 + r];
    b.y = W[(size_t)(k + 2 * hf + 1) * kH + n0 + r];
    // A fragments (16x4): same half-lane K split, M = lane%16
    v2f a0 = *(const v2f*)&xs[(r) * LDX + k + 2 * hf];
    v2f a1 = *(const v2f*)&xs[(16 + r) * LDX + k + 2 * hf];
    acc0 = __builtin_amdgcn_wmma_f32_16x16x4_f32(false, a0, false, b, (short)0,
                                                 acc0, false, false);
    acc1 = __builtin_amdgcn_wmma_f32_16x16x4_f32(false, a1, false, b, (short)0,
                                                 acc1, false, false);
  }

  // C/D layout: VGPR i -> M = i (lanes 0-15) / i+8 (lanes 16-31); N = lane%16
  const int col = n0 + r;
#pragma unroll
  for (int i = 0; i < 8; ++i) {
    int row0 = rb + i + hf * 8;
    h[(size_t)row0 * kH + col]        = acc0[i];
    h[(size_t)(row0 + 16) * kH + col] = acc1[i];
  }
}

// ---------------- aggregation: self-loop init, then edge scatter -------------
__global__ void k_agg_init(const float* __restrict__ h, const float* __restrict__ dis,
                           float* __restrict__ agg) {
  int t = blockIdx.x * blockDim.x + threadIdx.x;  // over N*H/4
  if (t >= kN * kH / 4) return;
  int i  = t / (kH / 4);
  int f  = (t % (kH / 4)) * 4;
  float d = dis[i];
  float w = d * d;  // norm for the self-loop edge (i -> i)
  float4 v = *(const float4*)(h + (size_t)i * kH + f);
  float4 o = make_float4(v.x * w, v.y * w, v.z * w, v.w * w);
  *(float4*)(agg + (size_t)i * kH + f) = o;
}

__global__ void k_edge_scatter(const int* __restrict__ ei, const float* __restrict__ h,
                               const float* __restrict__ dis, float* __restrict__ agg) {
  int t = blockIdx.x * blockDim.x + threadIdx.x;  // E * 32 threads
  int e = t >> 5;
  int f = (t & 31) * 4;
  int src = ei[e];
  int dst = ei[kE + e];
  float nrm = dis[src] * dis[dst];
  float4 v = *(const float4*)(h + (size_t)src * kH + f);
  float* a = agg + (size_t)dst * kH + f;
  atomAddF(a + 0, v.x * nrm);
  atomAddF(a + 1, v.y * nrm);
  atomAddF(a + 2, v.z * nrm);
  atomAddF(a + 3, v.w * nrm);
}

// ---------------- head: bias + ReLU + BayesLinear + log_softmax --------------
__global__ __launch_bounds__(256) void k_head(
    const float* __restrict__ agg, const float* __restrict__ gcn_b,
    const float* __restrict__ w_mu, const float* __restrict__ w_ls,
    const float* __restrict__ b_mu, const float* __restrict__ b_ls,
    const float* __restrict__ eps_w, const float* __restrict__ eps_b,
    float* __restrict__ out) {
  __shared__ float wsm[kC * kH];
  __shared__ float bsm[kC];
  __shared__ float gbs[kH];
  int tid = threadIdx.x;
  for (int i = tid; i < kC * kH; i += 256)
    wsm[i] = w_mu[i] + __expf(w_ls[i]) * eps_w[i];
  if (tid < kC)  bsm[tid] = b_mu[tid] + __expf(b_ls[tid]) * eps_b[tid];
  if (tid < kH)  gbs[tid] = gcn_b[tid];
  __syncthreads();

  int n = blockIdx.x * blockDim.x + tid;
  if (n >= kN) return;

  float logit[kC];
#pragma unroll
  for (int c = 0; c < kC; ++c) logit[c] = bsm[c];

  const float* arow = agg + (size_t)n * kH;
#pragma unroll 2
  for (int k = 0; k < kH; k += 4) {
    float4 av = *(const float4*)(arow + k);
    float a0 = fmaxf(av.x + gbs[k + 0], 0.0f);
    float a1 = fmaxf(av.y + gbs[k + 1], 0.0f);
    float a2 = fmaxf(av.z + gbs[k + 2], 0.0f);
    float a3 = fmaxf(av.w + gbs[k + 3], 0.0f);
#pragma unroll
    for (int c = 0; c < kC; ++c) {
      const float* wr = &wsm[c * kH + k];
      logit[c] += a0 * wr[0] + a1 * wr[1] + a2 * wr[2] + a3 * wr[3];
    }
  }

  float m = logit[0];
#pragma unroll
  for (int c = 1; c < kC; ++c) m = fmaxf(m, logit[c]);
  float s = 0.0f;
#pragma unroll
  for (int c = 0; c < kC; ++c) s += __expf(logit[c] - m);
  float lse = m + __logf(s);
  float* orow = out + (size_t)n * kC;
#pragma unroll
  for (int c = 0; c < kC; ++c) orow[c] = logit[c] - lse;
}

// ---------------- launcher ---------------------------------------------------
extern "C" void kernel_launch(void* const* d_in, const int* in_sizes, int n_in,
                              void* d_out, int out_size, void* d_ws, size_t ws_size,
                              hipStream_t stream) {
  const float* x    = (const float*)d_in[0];
  const int*   ei   = (const int*)d_in[1];
  const float* W    = (const float*)d_in[2];
  const float* gcnb = (const float*)d_in[3];
  const float* wmu  = (const float*)d_in[4];
  const float* wls  = (const float*)d_in[5];
  const float* bmu  = (const float*)d_in[6];
  const float* bls  = (const float*)d_in[7];
  const float* epsw = (const float*)d_in[8];
  const float* epsb = (const float*)d_in[9];
  float* out = (float*)d_out;

  // workspace layout: dis[N] | h[N*H] | agg[N*H]   (~103 MB)
  char* ws = (char*)d_ws;
  size_t o1 = ((size_t)kN * sizeof(float) + 255) & ~(size_t)255;
  size_t o2 = o1 + (size_t)kN * kH * sizeof(float);
  float* dis = (float*)ws;
  float* h   = (float*)(ws + o1);
  float* agg = (float*)(ws + o2);

  k_deg_init   <<<(kN + 255) / 256, 256, 0, stream>>>(dis);
  k_deg_count  <<<(kE + 255) / 256, 256, 0, stream>>>(ei, dis);
  k_deg_rsqrt  <<<(kN + 255) / 256, 256, 0, stream>>>(dis);
  k_gemm_xw    <<<kN / MT, 256, 0, stream>>>(x, W, h);
  k_agg_init   <<<(kN * kH / 4) / 256, 256, 0, stream>>>(h, dis, agg);
  k_edge_scatter<<<(kE * 32) / 256, 256, 0, stream>>>(ei, h, dis, agg);
  k_head       <<<(kN + 255) / 256, 256, 0, stream>>>(agg, gcnb, wmu, wls,
                                                      bmu, bls, epsw, epsb, out);
}